// BoxQueryAndGroup_73787538145796
// MI455X (gfx1250) — compile-verified
//
#include <hip/hip_runtime.h>

typedef __attribute__((ext_vector_type(4))) float v4f;   // native vector type

// Problem constants (match reference)
#define B_    4
#define N_    16384
#define NQ_   1024
#define C_    128
#define NS_   64
#define R2_   0.09f
#define TILE_ 2048   // points per LDS tile (24 KB)
#define QPB_  8      // queries per block (one wave each)
#define CHUNKS_ ((TILE_ * 3 * 4) / (16 * 256))   // b128 async loads per thread = 6

// ---------------------------------------------------------------------------
// Kernel A: box/ball selection.  One wave (32 lanes) per query.
// key_xyz[b] is streamed through LDS with a DOUBLE-BUFFERED async
// global->LDS pipeline (CDNA5 ASYNCcnt): while the 8 waves scan tile t in
// buffer tb, the 6 async b128 loads for tile t+1 are in flight into the other
// buffer.  Async loads complete in order, so `s_wait_asynccnt 6` guarantees
// the previous group of 6 has landed.
// Ordered compaction via ballot + prefix popcount reproduces the stable
// argsort of the 0/1 selection key (selected indices in ascending order).
// ---------------------------------------------------------------------------
__global__ __launch_bounds__(256) void box_select_kernel(
    const float* __restrict__ key_xyz,    // (B, N, 3)
    const float* __restrict__ query_xyz,  // (B, NQ, 6)
    float* __restrict__ out_xyz,          // (B, 3, NQ, NS)
    float* __restrict__ out_mask,         // (B, NQ, NS)  0.0/1.0
    int*   __restrict__ idx_ws)           // (B, NQ, NS)
{
  __shared__ __align__(16) float s_tile[2][TILE_ * 3];    // 2 x 24 KB
  __shared__ unsigned short s_in[QPB_][NS_];              // 1 KB
  __shared__ unsigned short s_bl[QPB_][NS_];              // 1 KB

  const int t    = threadIdx.x;
  const int lane = t & 31;
  const int w    = t >> 5;
  const int b    = blockIdx.y;
  const int q    = blockIdx.x * QPB_ + w;

  const float* qp = query_xyz + ((size_t)b * NQ_ + q) * 6;
  const float cx = qp[0], cy = qp[1], cz = qp[2];
  const float hx = 0.5f * qp[3], hy = 0.5f * qp[4], hz = 0.5f * qp[5];

  int icount = 0;   // total inside-box points (uniform across wave)
  int bcount = 0;   // total ball points

  // LDS byte address of tile buffers (flat-pointer low 32 bits == LDS offset)
  const unsigned lds0 =
      (unsigned)(unsigned long long)(uintptr_t)(void*)&s_tile[0][0];
  const unsigned long long gbase =
      (unsigned long long)(uintptr_t)(key_xyz + (size_t)b * N_ * 3);

  // issue the 6 per-thread async b128 loads for one tile into buffer `buf`
  auto issue_tile = [&](int tile, int buf) {
    const unsigned long long gsrc = gbase + (unsigned long long)tile * 12ull;
    const unsigned ldsb = lds0 + (unsigned)buf * (TILE_ * 3 * 4);
    #pragma unroll
    for (int i = 0; i < CHUNKS_; ++i) {
      const unsigned off = (unsigned)((i * 256 + t) * 16);
      const unsigned lds_addr = ldsb + off;
      const unsigned long long ga = gsrc + (unsigned long long)off;
      asm volatile("global_load_async_to_lds_b128 %0, %1, off"
                   :: "v"(lds_addr), "v"(ga) : "memory");
    }
  };

  issue_tile(0, 0);   // prologue: prime the pipeline

  int tb = 0;
  for (int tile = 0; tile < N_; tile += TILE_, tb ^= 1) {
    if (tile + TILE_ < N_) {
      issue_tile(tile + TILE_, tb ^ 1);                 // prefetch next tile
      asm volatile("s_wait_asynccnt 6" ::: "memory");   // tile `tile` landed
    } else {
      asm volatile("s_wait_asynccnt 0" ::: "memory");
    }
    __syncthreads();   // all waves' slices of this tile are in LDS

    const float* __restrict__ tp = s_tile[tb];
    for (int n0 = 0; n0 < TILE_; n0 += 32) {
      const int p = n0 + lane;
      const float x = tp[3 * p + 0];
      const float y = tp[3 * p + 1];
      const float z = tp[3 * p + 2];
      // explicit RN ops (no fma contraction) to bit-match the reference
      const float dx = fabsf(__fsub_rn(x, cx));
      const float dy = fabsf(__fsub_rn(y, cy));
      const float dz = fabsf(__fsub_rn(z, cz));
      const bool inside = (dx <= hx) && (dy <= hy) && (dz <= hz);
      const float d2 = __fadd_rn(
          __fadd_rn(__fmul_rn(dx, dx), __fmul_rn(dy, dy)), __fmul_rn(dz, dz));
      const bool ball = d2 <= R2_;

      const unsigned mi = (unsigned)__ballot(inside);
      const unsigned mb = (unsigned)__ballot(ball);
      const unsigned below = (1u << lane) - 1u;

      const int pi = icount + __popc(mi & below);
      if (inside && pi < NS_) s_in[w][pi] = (unsigned short)(tile + p);
      icount += __popc(mi);

      const int pb = bcount + __popc(mb & below);
      if (ball && pb < NS_) s_bl[w][pb] = (unsigned short)(tile + p);
      bcount += __popc(mb);
    }
    __syncthreads();   // everyone done reading buf tb before it is refilled
  }

  // --- emit idx, mask, grouped_xyz for this query ---
  const int total = (icount > 0) ? icount : bcount;
  const int cnt   = (total < NS_) ? total : NS_;
  const unsigned short* list = (icount > 0) ? s_in[w] : s_bl[w];
  const size_t qoff = ((size_t)b * NQ_ + q) * NS_;

  for (int s = lane; s < NS_; s += 32) {
    const int p = (s < cnt) ? (int)list[s] : 0;
    idx_ws[qoff + s] = p;
    __builtin_nontemporal_store((s >= cnt && s != 0) ? 1.0f : 0.0f,
                                &out_mask[qoff + s]);
    const float* kp = key_xyz + ((size_t)b * N_ + p) * 3;
    __builtin_nontemporal_store(__fsub_rn(kp[0], cx),
        &out_xyz[(((size_t)b * 3 + 0) * NQ_ + q) * NS_ + s]);
    __builtin_nontemporal_store(__fsub_rn(kp[1], cy),
        &out_xyz[(((size_t)b * 3 + 1) * NQ_ + q) * NS_ + s]);
    __builtin_nontemporal_store(__fsub_rn(kp[2], cz),
        &out_xyz[(((size_t)b * 3 + 2) * NQ_ + q) * NS_ + s]);
  }
}

// ---------------------------------------------------------------------------
// Kernel B: feature gather.  One block per (b, q).  key_features (33.5 MB) is
// L2-resident; the 134 MB output is streamed with nontemporal b128 stores so
// it doesn't evict the feature table from L2.  Each thread owns 4 consecutive
// `s` slots -> one global_store_b128 (NT) per 16 output bytes.
// ---------------------------------------------------------------------------
__global__ __launch_bounds__(256) void gather_feat_kernel(
    const float* __restrict__ key_feat,   // (B, C, N)
    const int*   __restrict__ idx_ws,     // (B, NQ, NS)
    float* __restrict__ out_feat)         // (B, C, NQ, NS)
{
  __shared__ int s_idx[NS_];
  const int q = blockIdx.x;
  const int b = blockIdx.y;
  const int t = threadIdx.x;

  if (t < NS_) s_idx[t] = idx_ws[((size_t)b * NQ_ + q) * NS_ + t];
  __syncthreads();

  const int s4 = (t & 15) * 4;             // 4 consecutive s slots
  const int c0 = t >> 4;                   // 0..15
  const int p0 = s_idx[s4 + 0];
  const int p1 = s_idx[s4 + 1];
  const int p2 = s_idx[s4 + 2];
  const int p3 = s_idx[s4 + 3];

  const float* fb = key_feat + (size_t)b * C_ * N_;
  float* ob = out_feat + (((size_t)b * C_) * NQ_ + q) * NS_ + s4;

  #pragma unroll 2
  for (int c = c0; c < C_; c += 16) {
    const float* fr = fb + (size_t)c * N_;
    v4f v;
    v.x = fr[p0];
    v.y = fr[p1];
    v.z = fr[p2];
    v.w = fr[p3];
    __builtin_nontemporal_store(v, (v4f*)(ob + (size_t)c * NQ_ * NS_));
  }
}

// ---------------------------------------------------------------------------
extern "C" void kernel_launch(void* const* d_in, const int* in_sizes, int n_in,
                              void* d_out, int out_size, void* d_ws,
                              size_t ws_size, hipStream_t stream) {
  const float* key_xyz  = (const float*)d_in[0];  // (B,N,3)
  const float* key_feat = (const float*)d_in[1];  // (B,C,N)
  const float* query    = (const float*)d_in[2];  // (B,NQ,6)

  float* out0 = (float*)d_out;                                   // grouped_xyz
  float* out1 = out0 + (size_t)B_ * 3 * NQ_ * NS_;               // grouped_feat
  float* out2 = out1 + (size_t)B_ * C_ * NQ_ * NS_;              // mask

  int* idx_ws = (int*)d_ws;   // B*NQ*NS ints = 1 MB

  dim3 gA(NQ_ / QPB_, B_);
  box_select_kernel<<<gA, 256, 0, stream>>>(key_xyz, query, out0, out2, idx_ws);

  dim3 gB(NQ_, B_);
  gather_feat_kernel<<<gB, 256, 0, stream>>>(key_feat, idx_ws, out1);
}